// SplineCoefs_from_GriddedData_85667417686559
// MI455X (gfx1250) — compile-verified
//
#include <hip/hip_runtime.h>

// ---------------------------------------------------------------------------
// Natural cubic spline coefficient sweep, 257^3 -> 259^3, as 3 batched GEMMs
//   out(259 x M) = H(259 x 257) @ y(257 x M)   per axis
// using V_WMMA_F32_16X16X4_F32 (fp32 matrix core, matches reference fp32).
// Roofline: ~27 GFLOP, ~0.4 GB HBM traffic -> both tens of us on MI455X.
// Each wave owns a full 259-row output strip (17 accumulators), so X is
// streamed exactly once per sweep; H (~283 KB) stays hot in L2/WGP$.
// ---------------------------------------------------------------------------

typedef __attribute__((ext_vector_type(2))) float v2f;
typedef __attribute__((ext_vector_type(8))) float v8f;

#define LDH   260   // padded H cols (mult of 4); cols 257..259 are zero
#define HROWS 272   // 17 tiles * 16 rows; rows 259..271 are zero
#define NT    17    // output-row tiles of 16 covering 259

static __device__ __forceinline__ v8f wmma4(v2f a, v2f b, v8f c) {
  // D = A(16x4,f32) * B(4x16,f32) + C(16x16,f32)
  return __builtin_amdgcn_wmma_f32_16x16x4_f32(false, a, false, b, (short)0, c,
                                               false, false);
}
static __device__ __forceinline__ int imin(int a, int b) { return a < b ? a : b; }

// ---------------------------------------------------------------------------
// Build padded operator HP (HROWS x LDH) in workspace.
// Thread j Thomas-solves tridiag(1,4,1)_{255} x = e_j (A symmetric => x is
// row j of Ainv), then writes H row 2+j with boundary folds. Rows 0/1/257/258
// are the closed-form boundary rows. All padding zero.
// ---------------------------------------------------------------------------
__global__ void build_H(float* __restrict__ HP) {
  const int t = threadIdx.x;  // 256 threads, single block
  for (int i = t; i < HROWS * LDH; i += 256) HP[i] = 0.0f;
  __syncthreads();

  __shared__ float x0s[255], x254s[255];

  if (t < 255) {
    double cp[255], dp[255], x[255];
    cp[0] = 0.25;
    dp[0] = (t == 0) ? 0.25 : 0.0;
    for (int i = 1; i < 255; ++i) {
      double denom = 4.0 - cp[i - 1];
      cp[i] = 1.0 / denom;
      double Bi = (i == t) ? 1.0 : 0.0;
      dp[i] = (Bi - dp[i - 1]) / denom;
    }
    x[254] = dp[254];
    for (int i = 253; i >= 0; --i) x[i] = dp[i] - cp[i] * x[i + 1];

    float* row = HP + (size_t)(2 + t) * LDH;
    row[0] = (float)(-x[0] / 6.0);
    for (int c = 1; c <= 255; ++c) row[c] = (float)x[c - 1];
    row[256] = (float)(-x[254] / 6.0);

    if (t == 0)   for (int i = 0; i < 255; ++i) x0s[i]   = (float)x[i];
    if (t == 254) for (int i = 0; i < 255; ++i) x254s[i] = (float)x[i];
  }
  __syncthreads();

  // Boundary rows. G[j,c]: c==0 -> -Ainv[j,0]/6 ; 1..255 -> Ainv[j,c-1] ;
  // c==256 -> -Ainv[j,254]/6.  row0 = (1/3)e0 - G[0,:], row258 = (1/3)e256 - G[254,:]
  for (int c = t; c < 257; c += 256) {
    float g0   = (c == 0) ? (-x0s[0] / 6.0f)
               : (c <= 255 ? x0s[c - 1] : (-x0s[254] / 6.0f));
    float g254 = (c == 0) ? (-x254s[0] / 6.0f)
               : (c <= 255 ? x254s[c - 1] : (-x254s[254] / 6.0f));
    HP[0 * LDH + c]   = ((c == 0)   ? (1.0f / 3.0f) : 0.0f) - g0;
    HP[258 * LDH + c] = ((c == 256) ? (1.0f / 3.0f) : 0.0f) - g254;
  }
  if (t == 0) {
    HP[1 * LDH + 0]     = 1.0f / 6.0f;  // row 1  = (1/6) e_0
    HP[257 * LDH + 256] = 1.0f / 6.0f;  // row 257= (1/6) e_256
  }
}

// ---------------------------------------------------------------------------
// Axes 0 & 1 (sweep axis NOT innermost): C[i',m] = sum_k H[i',k] * X[k*S + m].
// One wave per 16-column strip. Main K-loop is clamp-free (pure pointer
// increments); the K=256..259 tail is peeled (H cols 257..259 are zero).
// ---------------------------------------------------------------------------
__global__ __launch_bounds__(32)
void spline_rows(const float* __restrict__ X, float* __restrict__ out,
                 const float* __restrict__ HP,
                 int M, long long strideK, long long outRowStride,
                 long long batchInStride, long long batchOutStride) {
  const int lane = threadIdx.x;
  const int half = lane >> 4;      // 0: K={k,k+1}, 1: K={k+2,k+3}
  const int l16  = lane & 15;
  X   += (long long)blockIdx.y * batchInStride;
  out += (long long)blockIdx.y * batchOutStride;

  const int m  = blockIdx.x * 16 + l16;
  const int mc = imin(m, M - 1);   // clamp partial tile (stores are guarded)

  v8f acc[NT];
#pragma unroll
  for (int t = 0; t < NT; ++t) acc[t] = (v8f){0, 0, 0, 0, 0, 0, 0, 0};

  const long long sK = strideK;
  const float* bp = X + (long long)(half * 2) * sK + mc;  // B frag base
  const float* hb = HP + (size_t)l16 * LDH + half * 2;    // A frag base

  for (int k = 0; k < 256; k += 4) {   // kb = k+half*2 <= 254, kb+1 <= 255
    v2f b;  // B 4x16: V0 lanes0-15 K=k, lanes16-31 K=k+2; V1: k+1 / k+3
    b.x = bp[0];
    b.y = bp[sK];
#pragma unroll
    for (int t = 0; t < NT; ++t) {
      v2f a;  // A 16x4: lanes0-15 rows, V0/V1 = K {kb, kb+1}
      a.x = hb[(size_t)t * 16 * LDH + k];
      a.y = hb[(size_t)t * 16 * LDH + k + 1];
      acc[t] = wmma4(a, b, acc[t]);
    }
    bp += 4 * sK;
  }
  {  // tail K=256..259: X clamped to plane 256; H cols 257..259 are zero
    const float xv = X[256LL * sK + mc];
    v2f b;
    b.x = xv;
    b.y = xv;
#pragma unroll
    for (int t = 0; t < NT; ++t) {
      v2f a;
      a.x = hb[(size_t)t * 16 * LDH + 256];
      a.y = hb[(size_t)t * 16 * LDH + 257];
      acc[t] = wmma4(a, b, acc[t]);
    }
  }

  if (m < M) {
#pragma unroll
    for (int t = 0; t < NT; ++t) {
#pragma unroll
      for (int j = 0; j < 8; ++j) {   // D: VGPR j -> row j (lanes<16) / j+8
        const int r = t * 16 + half * 8 + j;
        if (r < 259) out[(long long)r * outRowStride + m] = acc[t][j];
      }
    }
  }
}

// ---------------------------------------------------------------------------
// Axis 2 (sweep axis innermost): C(M x 259) = X(M x 257, row-major) @ H^T.
// X rows are contiguous in k -> A-frag loads are adjacent per lane.
// ---------------------------------------------------------------------------
__global__ __launch_bounds__(32)
void spline_cols(const float* __restrict__ X, float* __restrict__ out,
                 const float* __restrict__ HP, int Mrows) {
  const int lane = threadIdx.x;
  const int half = lane >> 4;
  const int l16  = lane & 15;

  const int mr  = blockIdx.x * 16 + l16;
  const int mrc = imin(mr, Mrows - 1);
  const float* xr = X + (long long)mrc * 257 + half * 2;
  const float* hb = HP + (size_t)l16 * LDH + half * 2;

  v8f acc[NT];
#pragma unroll
  for (int t = 0; t < NT; ++t) acc[t] = (v8f){0, 0, 0, 0, 0, 0, 0, 0};

  for (int k = 0; k < 256; k += 4) {
    v2f a;  // A = X tile (16 m-rows x 4 k)
    a.x = xr[k];
    a.y = xr[k + 1];
#pragma unroll
    for (int t = 0; t < NT; ++t) {
      // B[k,n] = H^T[k,n] = HP[n*LDH + k], n = t*16 + l16
      v2f b;
      b.x = hb[(size_t)t * 16 * LDH + k];
      b.y = hb[(size_t)t * 16 * LDH + k + 1];
      acc[t] = wmma4(a, b, acc[t]);
    }
  }
  {  // tail K=256..259: X clamped to col 256; H cols 257..259 are zero
    const float xv = X[(long long)mrc * 257 + 256];
    v2f a;
    a.x = xv;
    a.y = xv;
#pragma unroll
    for (int t = 0; t < NT; ++t) {
      v2f b;
      b.x = hb[(size_t)t * 16 * LDH + 256];
      b.y = hb[(size_t)t * 16 * LDH + 257];
      acc[t] = wmma4(a, b, acc[t]);
    }
  }

#pragma unroll
  for (int t = 0; t < NT; ++t) {
    const int col = t * 16 + l16;     // i' dimension (259)
    if (col < 259) {
#pragma unroll
      for (int j = 0; j < 8; ++j) {
        const int r = blockIdx.x * 16 + half * 8 + j;  // m dimension
        if (r < Mrows) out[(long long)r * 259 + col] = acc[t][j];
      }
    }
  }
}

// ---------------------------------------------------------------------------
// Orchestration. Buffers:
//   HP (272x260 f32)          @ d_ws + 0        (282,880 B, 128-aligned)
//   T2 (259*259*257 f32)      @ d_ws + 282,880  (~69 MB)
//   T1 (259*257*257 f32) lives in d_out (fits: 259*257^2 < 259^3)
// axis0: d_in -> d_out(T1); axis1: d_out -> T2; axis2: T2 -> d_out (final)
// ---------------------------------------------------------------------------
extern "C" void kernel_launch(void* const* d_in, const int* in_sizes, int n_in,
                              void* d_out, int out_size, void* d_ws,
                              size_t ws_size, hipStream_t stream) {
  const float* y = (const float*)d_in[0];
  float* out = (float*)d_out;
  float* HP  = (float*)d_ws;
  float* T2  = (float*)((char*)d_ws + 282880);

  hipLaunchKernelGGL(build_H, dim3(1), dim3(256), 0, stream, HP);

  // axis 0: (257,257,257) -> (259,257,257), M = 257*257 = 66049, stride 66049
  hipLaunchKernelGGL(spline_rows, dim3(4129, 1), dim3(32), 0, stream,
                     y, out, HP, 66049, 66049LL, 66049LL, 0LL, 0LL);

  // axis 1: (259,257,257) -> (259,259,257), batched over d0=259
  hipLaunchKernelGGL(spline_rows, dim3(17, 259), dim3(32), 0, stream,
                     out, T2, HP, 257, 257LL, 257LL, 66049LL, 66563LL);

  // axis 2: (259,259,257) -> (259,259,259), M rows = 259*259 = 67081
  hipLaunchKernelGGL(spline_cols, dim3(4193, 1), dim3(32), 0, stream,
                     T2, out, HP, 67081);
}